// AttnDecoderRNN_14869176778972
// MI455X (gfx1250) — compile-verified
//
#include <hip/hip_runtime.h>

typedef __attribute__((ext_vector_type(2))) float v2f;
typedef __attribute__((ext_vector_type(8))) float v8f;

// ---------------- problem dims ----------------
// H=1024, A=1024, O=8000, M=500, L=15
#define Hh 1024
#define Oo 8000
#define Mm 500
#define Ll 15

// ---------------- workspace layout (floats) ----------------
constexpr int WS_HPAD = 0;                 // 16x1024 padded hidden
constexpr int WS_EPAD = 16384;             // 16x2048 padded encoder_out
constexpr int WS_RLIN = 49152;             // 16x1024  hidden @ Ur^T
constexpr int WS_ZLIN = 65536;             // 16x1024  hidden @ Uz^T
constexpr int WS_T1   = 81920;             // 16x1024  hidden @ Uo^T (N=1000 used)
constexpr int WS_EZ   = 98304;             // 16x1024  enc @ Cz^T
constexpr int WS_ES   = 114688;            // 16x1024  enc @ Cs^T
constexpr int WS_ECO  = 131072;            // 16x1024  enc @ Co^T (N=1000 used)
constexpr int WS_RH   = 147456;            // 16x1024  r * hidden (row15 = 0)
constexpr int WS_SLIN = 163840;            // 16x1024  rh @ Us^T
constexpr int WS_ZBUF = 180224;            // 16x1024  z gate
constexpr int WS_YWR  = 196608;            // 1024 y @ Wr^T
constexpr int WS_YWZ  = 197632;            // 1024
constexpr int WS_YWS  = 198656;            // 1024
constexpr int WS_YVO  = 199680;            // 1024 (1000 used)
constexpr int WS_VALP = 200704;            // 1024 attention partials
constexpr int WS_ATT7 = 201728;            // 1 scalar (padded)
constexpr int WS_TI   = 201760;            // 512 (500 used)
constexpr int WS_WT   = 202272;            // 8000
constexpr int WS_LSE  = 210272;            // 8000
constexpr int WS_WTMM = 218272;            // 2 (wt max, wt min)
// total ~218274 floats (~0.84 MB)

// ---------------- output layout (floats) ----------------
constexpr long long OFF_OUT = 0;           // (1,8000,8000)
constexpr int OFF_S    = 64000000;         // s_i (15,1,1024)
constexpr int OFF_ATT  = 64015360;         // attention (8,)
constexpr int OFF_ENEW = 64015368;         // e_new (8,)

__device__ __forceinline__ float warp_sum(float v) {
  for (int o = 16; o > 0; o >>= 1) v += __shfl_down(v, o, 32);
  return v;
}

// ---------------- prep: pad A matrices to 16 rows ----------------
__global__ void k_prep(const float* __restrict__ hidden, const float* __restrict__ hf,
                       const float* __restrict__ hb, float* __restrict__ ws) {
  int idx = blockIdx.x * blockDim.x + threadIdx.x;
  if (idx < 16 * Hh) {                              // Hpad
    int m = idx >> 10, n = idx & 1023;
    ws[WS_HPAD + idx] = (m < Ll) ? hidden[m * Hh + n] : 0.f;
  } else if (idx < 16 * Hh + 16 * 2048) {           // Epad = concat(h_fwd, h_bwd)
    int t = idx - 16 * Hh;
    int m = t >> 11, c = t & 2047;
    float v = 0.f;
    if (m < Ll) v = (c < Hh) ? hf[m * Hh + c] : hb[m * Hh + (c - Hh)];
    ws[WS_EPAD + t] = v;
  } else if (idx < 16 * Hh + 16 * 2048 + Hh) {      // zero row 15 of rh
    int n = idx - (16 * Hh + 16 * 2048);
    ws[WS_RH + 15 * Hh + n] = 0.f;
  }
}

// ---------------- WMMA GEMM: C[16 x N] = A[16 x K] * W[N x K]^T ----------------
// One wave per 16-column tile; branchless column clamp keeps EXEC all-1s so the
// loop body is two global_load_b64 + one v_wmma_f32_16x16x4_f32.
__global__ void gemm16_wmma(const float* __restrict__ A, const float* __restrict__ W,
                            float* __restrict__ C, int K, int Nreal) {
  int wave = (blockIdx.x * blockDim.x + threadIdx.x) >> 5;
  int lane = threadIdx.x & 31;
  int ntiles = (Nreal + 15) >> 4;
  if (wave >= ntiles) return;                  // wave-uniform exit
  int n0   = wave << 4;
  int m    = lane & 15;                        // A row (both halves map M=0..15)
  int half = lane >> 4;                        // K sub-pair select
  int n    = n0 + (lane & 15);                 // B column
  int nc   = (n < Nreal) ? n : (Nreal - 1);    // clamp: garbage cols land in padded
                                               // C columns that are never read
  const float* Arow = A + m * K + 2 * half;
  const float* Brow = W + (size_t)nc * K + 2 * half;
  v8f acc = {};
#pragma unroll 4
  for (int k = 0; k < K; k += 4) {
    v2f a = *(const v2f*)(Arow + k);
    v2f b = *(const v2f*)(Brow + k);
    acc = __builtin_amdgcn_wmma_f32_16x16x4_f32(false, a, false, b, (short)0, acc,
                                                false, false);
  }
  // C/D layout: VGPR v -> row v + 8*half, lane&15 -> column
  for (int v = 0; v < 8; ++v)
    C[(v + 8 * half) * 1024 + n0 + (lane & 15)] = acc[v];
}

// ---------------- attention: per-n partial of Va * tanh(...) ----------------
__global__ void k_att1(const float* __restrict__ hidden, const float* __restrict__ hf,
                       const float* __restrict__ hb, const float* __restrict__ Wa,
                       const float* __restrict__ Wab, const float* __restrict__ Ua,
                       const float* __restrict__ Uab, const float* __restrict__ Va,
                       const int* __restrict__ cidx, float* __restrict__ ws) {
  int wave = (blockIdx.x * blockDim.x + threadIdx.x) >> 5;
  int lane = threadIdx.x & 31;
  if (wave >= Hh) return;
  int c0 = cidx[0];
  const float* hr = hidden + (size_t)c0 * Hh;
  const float* fr = hf + (size_t)c0 * Hh;
  const float* br = hb + (size_t)c0 * Hh;
  float s = 0.f;
  for (int k = lane; k < Hh; k += 32)   s += Wa[(size_t)wave * Hh + k] * hr[k];
  for (int k = lane; k < 2 * Hh; k += 32) {
    float hj = (k < Hh) ? fr[k] : br[k - Hh];
    s += Ua[(size_t)wave * 2 * Hh + k] * hj;
  }
  s = warp_sum(s);
  if (lane == 0) ws[WS_VALP + wave] = tanhf(s + Wab[wave] + Uab[wave]) * Va[wave];
}

// ---------------- attention reduce + softmax ----------------
__global__ void k_att2(const float* __restrict__ Vab, const float* __restrict__ evals,
                       float* __restrict__ ws, float* __restrict__ out) {
  __shared__ float red[256];
  int t = threadIdx.x;
  float s = 0.f;
  for (int i = t; i < Hh; i += 256) s += ws[WS_VALP + i];
  red[t] = s; __syncthreads();
  for (int o = 128; o > 0; o >>= 1) { if (t < o) red[t] += red[t + o]; __syncthreads(); }
  if (t == 0) {
    float val = red[0] + Vab[0];
    float e[8];
    for (int i = 0; i < 7; ++i) e[i] = evals[i];
    e[7] = val;
    float mx = e[0];
    for (int i = 1; i < 8; ++i) mx = fmaxf(mx, e[i]);
    float ex[8], den = 0.f;
    for (int i = 0; i < 8; ++i) { ex[i] = expf(e[i] - mx); den += ex[i]; }
    for (int i = 0; i < 8; ++i) { out[OFF_ENEW + i] = e[i]; out[OFF_ATT + i] = ex[i] / den; }
    ws[WS_ATT7] = ex[7] / den;
  }
}

// ---------------- y-GEMV: dst[n] = y . W[n,:], K=8000 ----------------
__global__ void k_ygemv(const float* __restrict__ embed, const int* __restrict__ tok,
                        const float* __restrict__ W, int Nreal, float* __restrict__ dst) {
  int wave = (blockIdx.x * blockDim.x + threadIdx.x) >> 5;
  int lane = threadIdx.x & 31;
  if (wave >= Nreal) return;
  const float* y = embed + (size_t)tok[0] * Oo;
  const float* wr = W + (size_t)wave * Oo;
  float s = 0.f;
  for (int k = lane; k < Oo; k += 32) s += y[k] * wr[k];
  s = warp_sum(s);
  if (lane == 0) dst[wave] = s;
}

// ---------------- gates r, z; build rh = r*hidden ----------------
__global__ void k_rz(const float* __restrict__ Wrb, const float* __restrict__ Urb,
                     const float* __restrict__ Wzb, const float* __restrict__ Uzb,
                     const float* __restrict__ Czb, float* __restrict__ ws) {
  int idx = blockIdx.x * blockDim.x + threadIdx.x;
  if (idx >= Ll * Hh) return;
  int n = idx & 1023;
  float att7 = ws[WS_ATT7];
  float r = ws[WS_YWR + n] + Wrb[n] + ws[WS_RLIN + idx] + Urb[n];
  r = 1.f / (1.f + expf(-r));
  float z = ws[WS_YWZ + n] + Wzb[n] + ws[WS_ZLIN + idx] + Uzb[n]
          + att7 * ws[WS_EZ + idx] + Czb[n];
  z = 1.f / (1.f + expf(-z));
  ws[WS_ZBUF + idx] = z;
  ws[WS_RH + idx] = r * ws[WS_HPAD + idx];
}

// ---------------- s_tilde, s_i ----------------
__global__ void k_s(const float* __restrict__ Wsb, const float* __restrict__ Usb,
                    const float* __restrict__ Csb, float* __restrict__ ws,
                    float* __restrict__ out) {
  int idx = blockIdx.x * blockDim.x + threadIdx.x;
  if (idx >= Ll * Hh) return;
  int n = idx & 1023;
  float att7 = ws[WS_ATT7];
  float st = tanhf(ws[WS_YWS + n] + Wsb[n] + ws[WS_SLIN + idx] + Usb[n]
                 + att7 * ws[WS_ES + idx] + Csb[n]);
  float h = ws[WS_HPAD + idx];
  float z = ws[WS_ZBUF + idx];
  out[OFF_S + idx] = (1.f - z) * h + z * st;
}

// ---------------- t_tilde row -> odd/even max -> t_i ----------------
__global__ void k_t(const float* __restrict__ Uob, const float* __restrict__ Vob,
                    const float* __restrict__ Cob, const int* __restrict__ cidx,
                    float* __restrict__ ws) {
  __shared__ float row[2 * Mm];
  int t = threadIdx.x;
  int c0 = cidx[0];
  float att7 = ws[WS_ATT7];
  if (t < 2 * Mm) {
    row[t] = ws[WS_T1 + c0 * 1024 + t] + Uob[t] + ws[WS_YVO + t] + Vob[t]
           + att7 * ws[WS_ECO + c0 * 1024 + t] + Cob[t];
  }
  __syncthreads();
  if (t < Mm) {
    float odd  = row[(2 * t + 2 * Mm - 1) % (2 * Mm)];
    float even = row[2 * t];
    ws[WS_TI + t] = fmaxf(odd, even);
  }
}

// ---------------- wt = Wo_w @ t_i ----------------
__global__ void k_wt(const float* __restrict__ Wo, float* __restrict__ ws) {
  int wave = (blockIdx.x * blockDim.x + threadIdx.x) >> 5;
  int lane = threadIdx.x & 31;
  if (wave >= Oo) return;
  const float* wr = Wo + (size_t)wave * Mm;
  float s = 0.f;
  for (int k = lane; k < Mm; k += 32) s += wr[k] * ws[WS_TI + k];
  s = warp_sum(s);
  if (lane == 0) ws[WS_WT + wave] = s;
}

// ---------------- min/max of wt (closed-form column max) ----------------
__global__ void k_minmax(float* __restrict__ ws) {
  __shared__ float mx[256], mn[256];
  int t = threadIdx.x;
  float a = -1e30f, b = 1e30f;
  for (int i = t; i < Oo; i += 256) {
    float v = ws[WS_WT + i];
    a = fmaxf(a, v); b = fminf(b, v);
  }
  mx[t] = a; mn[t] = b; __syncthreads();
  for (int o = 128; o > 0; o >>= 1) {
    if (t < o) { mx[t] = fmaxf(mx[t], mx[t + o]); mn[t] = fminf(mn[t], mn[t + o]); }
    __syncthreads();
  }
  if (t == 0) { ws[WS_WTMM] = mx[0]; ws[WS_WTMM + 1] = mn[0]; }
}

// ---------------- per-column logsumexp over i of exp(wt_i * y_j) ----------------
__global__ void k_lse(const float* __restrict__ embed, const int* __restrict__ tok,
                      float* __restrict__ ws) {
  int j = blockIdx.x * blockDim.x + threadIdx.x;
  if (j >= Oo) return;
  const float* y = embed + (size_t)tok[0] * Oo;
  float yj = y[j];
  float Mlog = (yj >= 0.f) ? yj * ws[WS_WTMM] : yj * ws[WS_WTMM + 1];
  float Umax = expf(Mlog);                 // = max_i exp(wt_i * y_j)
  float s = 0.f;
  for (int i = 0; i < Oo; ++i)
    s += expf(expf(ws[WS_WT + i] * yj) - Umax);
  ws[WS_LSE + j] = Umax + logf(s);
}

// ---------------- final 8000x8000 log_softmax output ----------------
__global__ void k_out(const float* __restrict__ embed, const int* __restrict__ tok,
                      const float* __restrict__ ws, float* __restrict__ out) {
  int j = blockIdx.x * blockDim.x + threadIdx.x;   // column, coalesced
  int i = blockIdx.y;                              // row
  if (j >= Oo) return;
  const float* y = embed + (size_t)tok[0] * Oo;
  float u = expf(ws[WS_WT + i] * y[j]);
  out[OFF_OUT + (long long)i * Oo + j] = u - ws[WS_LSE + j];
}

extern "C" void kernel_launch(void* const* d_in, const int* in_sizes, int n_in,
                              void* d_out, int out_size, void* d_ws, size_t ws_size,
                              hipStream_t stream) {
  (void)in_sizes; (void)n_in; (void)out_size; (void)ws_size;
  const int*   tok    = (const int*)d_in[0];
  const float* hidden = (const float*)d_in[1];
  const float* hf     = (const float*)d_in[2];
  const float* hb     = (const float*)d_in[3];
  const float* evals  = (const float*)d_in[4];
  const float* embed  = (const float*)d_in[5];
  const float* Wa_w   = (const float*)d_in[6];
  const float* Wa_b   = (const float*)d_in[7];
  const float* Va_w   = (const float*)d_in[8];
  const float* Va_b   = (const float*)d_in[9];
  const float* Ua_w   = (const float*)d_in[10];
  const float* Ua_b   = (const float*)d_in[11];
  const float* Ws_w   = (const float*)d_in[12];
  const float* Ws_b   = (const float*)d_in[13];
  const float* Wz_w   = (const float*)d_in[14];
  const float* Wz_b   = (const float*)d_in[15];
  const float* Wr_w   = (const float*)d_in[16];
  const float* Wr_b   = (const float*)d_in[17];
  const float* Us_w   = (const float*)d_in[18];
  const float* Us_b   = (const float*)d_in[19];
  const float* Uz_w   = (const float*)d_in[20];
  const float* Uz_b   = (const float*)d_in[21];
  const float* Ur_w   = (const float*)d_in[22];
  const float* Ur_b   = (const float*)d_in[23];
  const float* Cs_w   = (const float*)d_in[24];
  const float* Cs_b   = (const float*)d_in[25];
  const float* Cz_w   = (const float*)d_in[26];
  const float* Cz_b   = (const float*)d_in[27];
  const float* Wo_w   = (const float*)d_in[28];
  const float* Uo_w   = (const float*)d_in[29];
  const float* Uo_b   = (const float*)d_in[30];
  const float* Vo_w   = (const float*)d_in[31];
  const float* Vo_b   = (const float*)d_in[32];
  const float* Co_w   = (const float*)d_in[33];
  const float* Co_b   = (const float*)d_in[34];
  const int*   cidx   = (const int*)d_in[35];

  float* out = (float*)d_out;
  float* ws  = (float*)d_ws;

  // 1) pad A matrices
  k_prep<<<(16*Hh + 16*2048 + Hh + 255)/256, 256, 0, stream>>>(hidden, hf, hb, ws);

  // 2) attention scalar path
  k_att1<<<(Hh*32)/256, 256, 0, stream>>>(hidden, hf, hb, Wa_w, Wa_b, Ua_w, Ua_b,
                                          Va_w, cidx, ws);
  k_att2<<<1, 256, 0, stream>>>(Va_b, evals, ws, out);

  // 3) WMMA GEMMs independent of attention (c_i factored as att7 * (E @ C^T))
  gemm16_wmma<<<8, 256, 0, stream>>>(ws + WS_HPAD, Ur_w, ws + WS_RLIN, Hh,   1024);
  gemm16_wmma<<<8, 256, 0, stream>>>(ws + WS_HPAD, Uz_w, ws + WS_ZLIN, Hh,   1024);
  gemm16_wmma<<<8, 256, 0, stream>>>(ws + WS_HPAD, Uo_w, ws + WS_T1,   Hh,   2*Mm);
  gemm16_wmma<<<8, 256, 0, stream>>>(ws + WS_EPAD, Cz_w, ws + WS_EZ,   2*Hh, 1024);
  gemm16_wmma<<<8, 256, 0, stream>>>(ws + WS_EPAD, Cs_w, ws + WS_ES,   2*Hh, 1024);
  gemm16_wmma<<<8, 256, 0, stream>>>(ws + WS_EPAD, Co_w, ws + WS_ECO,  2*Hh, 2*Mm);

  // 4) y-based GEMVs (32 MB weight streams each)
  k_ygemv<<<(1024*32)/256, 256, 0, stream>>>(embed, tok, Wr_w, 1024, ws + WS_YWR);
  k_ygemv<<<(1024*32)/256, 256, 0, stream>>>(embed, tok, Wz_w, 1024, ws + WS_YWZ);
  k_ygemv<<<(1024*32)/256, 256, 0, stream>>>(embed, tok, Ws_w, 1024, ws + WS_YWS);
  k_ygemv<<<(1000*32 + 255)/256, 256, 0, stream>>>(embed, tok, Vo_w, 2*Mm, ws + WS_YVO);

  // 5) gates, serial Us GEMM, state update
  k_rz<<<(Ll*Hh + 255)/256, 256, 0, stream>>>(Wr_b, Ur_b, Wz_b, Uz_b, Cz_b, ws);
  gemm16_wmma<<<8, 256, 0, stream>>>(ws + WS_RH, Us_w, ws + WS_SLIN, Hh, 1024);
  k_s<<<(Ll*Hh + 255)/256, 256, 0, stream>>>(Ws_b, Us_b, Cs_b, ws, out);

  // 6) output head
  k_t<<<1, 1024, 0, stream>>>(Uo_b, Vo_b, Co_b, cidx, ws);
  k_wt<<<(Oo*32)/256, 256, 0, stream>>>(Wo_w, ws);
  k_minmax<<<1, 256, 0, stream>>>(ws);
  k_lse<<<(Oo + 255)/256, 256, 0, stream>>>(embed, tok, ws);
  dim3 og((Oo + 255)/256, Oo);
  k_out<<<og, 256, 0, stream>>>(embed, tok, ws, out);
}